// Derenderer3d_2851858284920
// MI455X (gfx1250) — compile-verified
//
#include <hip/hip_runtime.h>
#include <hip/hip_bf16.h>
#include <math.h>

#define RSZ   128
#define BATCH 128
#define NC    8
#define NV    16384
#define NK    64
#define MAXM  5            // batches packed per WMMA B (5*3 = 15 cols <= 16)
#define MAXG  32           // sum ceil(n_c/5) <= (128 + 8*4)/5 = 32
#define TILES 4            // vertex tiles (128 verts each) per block
#define GRP_OFF 2048       // int offset of group table in ws

// d_out float offsets (return-order concatenation)
#define OFF_MASKS  0u
#define OFF_VERTS  2097152u      // B*R*R
#define OFF_THETAS 8388608u
#define OFF_ALPHAS 8388736u
#define OFF_ROT    8388864u
#define OFF_SCALES 8389376u
#define OFF_DEPTHS 8389504u
#define OFF_C2D    8389632u
#define OFF_TRANS  8389888u
#define OFF_CLP    8390272u

typedef float v2f __attribute__((ext_vector_type(2)));
typedef float v8f __attribute__((ext_vector_type(8)));

#define PI_F  3.14159265358979323846f
#define TWOPI 6.28318530717958647692f

__global__ void zero_masks_kernel(float* __restrict__ out) {
    unsigned i = blockIdx.x * blockDim.x + threadIdx.x;
    if (i < OFF_VERTS) out[OFF_MASKS + i] = 0.0f;
}

// One thread per batch: all per-batch scalars + per-batch params, then thread 0
// builds the class-packed group table (deterministic counting grouping).
__global__ void setup_kernel(const float* __restrict__ roi,
                             const float* __restrict__ focals,
                             const float* __restrict__ td,
                             const float* __restrict__ t2,
                             const float* __restrict__ ls,
                             const float* __restrict__ ld,
                             const float* __restrict__ cp,
                             float* __restrict__ out,
                             float* __restrict__ ws) {
    __shared__ int s_cls[BATCH];
    int b = threadIdx.x;
    if (b < BATCH) {
        float r0 = roi[b*4+0], r1 = roi[b*4+1], r2 = roi[b*4+2], r3 = roi[b*4+3];
        float mx = 0.5f*(r2 + r0), my = 0.5f*(r3 + r1);
        float dx = r2 - r0,        dy = r3 - r1;

        float theta = atan2f(td[b*2+1], td[b*2+0]);
        float qw = cosf(0.5f*theta), qy = sinf(0.5f*theta);

        float area  = dx * dy;
        float scale = expf(ls[b]);
        float depth = sqrtf(expf(ld[b]) / area);

        float cx = mx + t2[b*2+0]*dx;
        float cy = my + t2[b*2+1]*dy;

        float tux = cy, tuy = -cx;
        float inv = rsqrtf(tux*tux + tuy*tuy + 1.0f);
        float tx = depth*tux*inv, ty = depth*tuy*inv, tz = depth*(-1.0f)*inv;

        float alpha = -(theta - atanf(tx / tz));
        float a = alpha + PI_F;                 // python-mod into [0, 2pi)
        a = a - floorf(a / TWOPI) * TWOPI;
        alpha = a - PI_F;

        int   ci = 0;
        float cm = cp[b*NC];
        #pragma unroll
        for (int j = 1; j < NC; ++j) {
            float v = cp[b*NC + j];
            if (v > cm) { cm = v; ci = j; }
        }
        s_cls[b] = ci;

        out[OFF_THETAS + b] = theta;
        out[OFF_ALPHAS + b] = alpha;
        out[OFF_ROT + b*4+0] = qw;  out[OFF_ROT + b*4+1] = 0.0f;
        out[OFF_ROT + b*4+2] = qy;  out[OFF_ROT + b*4+3] = 0.0f;
        out[OFF_SCALES + b] = scale;
        out[OFF_DEPTHS + b] = depth;
        out[OFF_C2D + b*2+0] = cx;  out[OFF_C2D + b*2+1] = cy;
        out[OFF_TRANS + b*3+0] = tx; out[OFF_TRANS + b*3+1] = ty; out[OFF_TRANS + b*3+2] = tz;
        out[OFF_CLP + b] = logf(cm);

        ((int*)ws)[b*16 + 0] = ci;
        ws[b*16 + 1] = scale;
        ws[b*16 + 2] = qw;
        ws[b*16 + 3] = qy;
        ws[b*16 + 4] = tx;
        ws[b*16 + 5] = ty;
        ws[b*16 + 6] = tz;
        ws[b*16 + 7] = focals[b] * (1.0f / 64.0f);   // 2*focal/R
    }
    __syncthreads();
    if (b == 0) {
        int* gt = (int*)ws + GRP_OFF;
        int ng = 0;
        for (int c = 0; c < NC; ++c) {
            int cnt = 0, cur[MAXM];
            for (int q = 0; q < BATCH; ++q) {
                if (s_cls[q] == c) {
                    cur[cnt++] = q;
                    if (cnt == MAXM) {
                        int* e = gt + 8 + ng*8;
                        e[0] = c; e[1] = MAXM;
                        for (int u = 0; u < MAXM; ++u) e[2+u] = cur[u];
                        ng++; cnt = 0;
                    }
                }
            }
            if (cnt > 0) {
                int* e = gt + 8 + ng*8;
                e[0] = c; e[1] = cnt;
                for (int u = 0; u < MAXM; ++u) e[2+u] = (u < cnt) ? cur[u] : 0;
                ng++;
            }
        }
        gt[0] = ng;
    }
}

// Block: 256 threads = 8 waves. One group (<=5 same-class batches) per grid.y.
// Each wave: 16-row A tile of ffd_basis[cls], B = packed 15-column coeff block,
// v_wmma_f32_16x16x4_f32 x16 per tile, TILES tiles per block.
__global__ void __launch_bounds__(256)
vert_kernel(const float* __restrict__ ffd_coeffs,
            const float* __restrict__ base_vertices,
            const float* __restrict__ ffd_basis,
            const float* __restrict__ ws,
            float* __restrict__ out) {
    __shared__ float shc[MAXM * NK * 3];    // member coeff blocks (K x 3 each)
    __shared__ float shp[MAXM * 8];         // member params (slot 0 = batch idx bits)
    __shared__ float shv[128 * 17];         // C tile re-shard, stride-17 padded

    const int* gt = (const int*)ws + GRP_OFF;
    const int  ng = gt[0];
    const int  g  = blockIdx.y;
    if (g >= ng) return;
    const int* grp = gt + 8 + g*8;
    const int  cls = grp[0];
    const int  m   = grp[1];
    const int  tid = threadIdx.x;

    // Stage member coeffs + params in LDS.
    for (int i = tid; i < m * NK * 3; i += 256) {
        int j = i / (NK*3);
        shc[i] = ffd_coeffs[((size_t)grp[2+j] * NC + cls) * (NK*3) + (i - j*(NK*3))];
    }
    if (tid < m * 8) {
        int j = tid >> 3, r = tid & 7;
        int bj = grp[2 + j];
        if (r == 0) ((int*)shp)[j*8] = bj;
        else        shp[j*8 + r] = ws[bj*16 + r];
    }
    __syncthreads();

    const int wave = tid >> 5;
    const int lane = tid & 31;
    const int half = lane >> 4;   // lanes 16-31: K+2/K+3 rows of A and B frags
    const int mrow = lane & 15;   // A: M row; B/C: column

    // Hoist loop-invariant B fragments: col = 3*member + coord, cols >= 3m zero.
    const int  jm = mrow / 3;
    const int  cc = mrow - jm * 3;
    const bool act = (mrow < 3 * m);
    v2f bfrag[16];
    #pragma unroll
    for (int k4 = 0; k4 < 16; ++k4) {
        const int ka = k4*4 + half*2;
        float b0 = 0.0f, b1 = 0.0f;
        if (act) {
            b0 = shc[jm*(NK*3) + ka*3 + cc];
            b1 = shc[jm*(NK*3) + (ka+1)*3 + cc];
        }
        v2f bb; bb.x = b0; bb.y = b1;
        bfrag[k4] = bb;
    }

    for (int t = 0; t < TILES; ++t) {
        const int tile0 = (blockIdx.x * TILES + t) * 128;
        const float* A = ffd_basis + ((size_t)cls * NV + tile0 + wave*16 + mrow) * NK;

        v8f acc = {};
        #pragma unroll
        for (int k4 = 0; k4 < 16; ++k4) {
            v2f av = *(const v2f*)(A + k4*4 + half*2);   // 8B-aligned global_load_b64
            acc = __builtin_amdgcn_wmma_f32_16x16x4_f32(
                false, av, false, bfrag[k4], (short)0, acc, false, false);
        }

        // C frag: VGPR i -> row (wave*16 + half*8 + i), col = mrow.
        #pragma unroll
        for (int i = 0; i < 8; ++i)
            shv[(wave*16 + half*8 + i)*17 + mrow] = acc[i];
        __syncthreads();

        // Epilogue: m*128 (member, vertex) work items over 256 threads.
        for (int idx = tid; idx < (m << 7); idx += 256) {
            const int j2 = idx >> 7;
            const int v  = idx & 127;
            const int bj = ((const int*)shp)[j2*8];
            const float scale = shp[j2*8+1];
            const float qw    = shp[j2*8+2];
            const float qy    = shp[j2*8+3];
            const float ttx   = shp[j2*8+4];
            const float tty   = shp[j2*8+5];
            const float ttz   = shp[j2*8+6];
            const float feff  = shp[j2*8+7];

            const int vg = tile0 + v;
            const float* bp = base_vertices + ((size_t)cls * NV + vg) * 3;
            float x = (bp[0] + shv[v*17 + j2*3 + 0]) * scale;
            float y = (bp[1] + shv[v*17 + j2*3 + 1]) * scale;
            float z = (bp[2] + shv[v*17 + j2*3 + 2]) * scale;

            // quat rotate, q = (qw,0,qy,0): t = 2*cross(qv,v); v' = v + qw*t + cross(qv,t)
            float t0  = 2.0f*qy*z, t2v = -2.0f*qy*x;
            float rx = x + qw*t0 + qy*t2v + ttx;
            float ry = y + tty;
            float rz = z + qw*t2v - qy*t0 + ttz;

            float zsafe = (rz > -1.0e-4f) ? -1.0e-4f : rz;
            float u  = rx / (-zsafe) * feff;
            float vv = ry / (-zsafe) * feff;
            float px = (u  * 0.5f + 0.5f) * (float)RSZ;
            float py = (vv * 0.5f + 0.5f) * (float)RSZ;

            const size_t vo = OFF_VERTS + ((size_t)bj * NV + vg) * 3;
            out[vo + 0] = px;
            out[vo + 1] = py;
            out[vo + 2] = -rz;

            int xi = (int)px; xi = xi < 0 ? 0 : (xi > RSZ-1 ? RSZ-1 : xi);
            int yi = (int)py; yi = yi < 0 ? 0 : (yi > RSZ-1 ? RSZ-1 : yi);
            out[OFF_MASKS + (size_t)bj * RSZ * RSZ + yi * RSZ + xi] = 1.0f; // constant store
        }
        __syncthreads();
    }
}

extern "C" void kernel_launch(void* const* d_in, const int* in_sizes, int n_in,
                              void* d_out, int out_size, void* d_ws, size_t ws_size,
                              hipStream_t stream) {
    const float* roi = (const float*)d_in[0];
    const float* foc = (const float*)d_in[1];
    const float* td  = (const float*)d_in[2];
    const float* t2  = (const float*)d_in[3];
    const float* ls  = (const float*)d_in[4];
    const float* ld  = (const float*)d_in[5];
    const float* cp  = (const float*)d_in[6];
    const float* fc  = (const float*)d_in[7];   // ffd_coeffs  (B,C,K,3)
    const float* bv  = (const float*)d_in[8];   // base_vertices (C,V,3)
    const float* fb  = (const float*)d_in[9];   // ffd_basis   (C,V,K)
    float* out = (float*)d_out;
    float* ws  = (float*)d_ws;

    zero_masks_kernel<<<OFF_VERTS / 256, 256, 0, stream>>>(out);
    setup_kernel<<<1, 128, 0, stream>>>(roi, foc, td, t2, ls, ld, cp, out, ws);
    dim3 grid(NV / (128 * TILES), MAXG);
    vert_kernel<<<grid, 256, 0, stream>>>(fc, bv, fb, ws, out);
}